// TtMoeLayer_83416854823717
// MI455X (gfx1250) — compile-verified
//
#include <hip/hip_runtime.h>
#include <hip/hip_bf16.h>
#include <math.h>

// ---- problem constants (B=32, S=1, H=4096, O=4096, E=8, TOP_K=2) ----
#define NTOK 32
#define HDIM 4096
#define ODIM 4096
#define NEXP 8

typedef __attribute__((ext_vector_type(2))) float v2f;
typedef __attribute__((ext_vector_type(4))) float v4f;
typedef __attribute__((ext_vector_type(8))) float v8f;

#if defined(__HIP_DEVICE_COMPILE__)
#if !__has_builtin(__builtin_amdgcn_wmma_f32_16x16x4_f32)
#error "missing __builtin_amdgcn_wmma_f32_16x16x4_f32 on this toolchain (device pass)"
#endif
#endif

// -------------------------------------------------------------------------
// Kernel 1: gate logits + top-2 softmax -> per-(token,expert) weight table
// grid = 32 blocks (one per token), 256 threads
// -------------------------------------------------------------------------
__global__ void gate_topk_kernel(const float* __restrict__ x,
                                 const float* __restrict__ gate_w,
                                 float* __restrict__ w_out) {
    const int b = blockIdx.x;
    __shared__ float logits[NEXP];
    if (threadIdx.x < NEXP) logits[threadIdx.x] = 0.0f;
    __syncthreads();

    float acc[NEXP];
#pragma unroll
    for (int e = 0; e < NEXP; ++e) acc[e] = 0.0f;

    const float* xr = x + (size_t)b * HDIM;
    for (int h = threadIdx.x; h < HDIM; h += blockDim.x) {
        const float xv = xr[h];
        const float* gw = gate_w + (size_t)h * NEXP;   // [H,E] row-major
#pragma unroll
        for (int e = 0; e < NEXP; ++e) acc[e] += xv * gw[e];
    }
#pragma unroll
    for (int e = 0; e < NEXP; ++e) atomicAdd(&logits[e], acc[e]);  // ds_add_f32
    __syncthreads();

    if (threadIdx.x == 0) {
        // top-2 (first-occurrence tie-break, like jax.lax.top_k)
        int i0 = 0; float v0 = logits[0];
        for (int e = 1; e < NEXP; ++e)
            if (logits[e] > v0) { v0 = logits[e]; i0 = e; }
        int i1 = -1; float v1 = -INFINITY;
        for (int e = 0; e < NEXP; ++e)
            if (e != i0 && logits[e] > v1) { v1 = logits[e]; i1 = e; }
        const float e0 = __expf(v0 - v0);
        const float e1 = __expf(v1 - v0);
        const float inv = 1.0f / (e0 + e1);
        float* wr = w_out + b * NEXP;
#pragma unroll
        for (int e = 0; e < NEXP; ++e) wr[e] = 0.0f;
        wr[i0] = e0 * inv;
        wr[i1] = e1 * inv;
    }
}

// -------------------------------------------------------------------------
// Kernel 2: zero-init the output accumulator
// -------------------------------------------------------------------------
__global__ void zero_out_kernel(float* __restrict__ out, int n) {
    int i = blockIdx.x * blockDim.x + threadIdx.x;
    if (i < n) out[i] = 0.0f;
}

// -------------------------------------------------------------------------
// Kernel 3: per-expert fp32 WMMA GEMM, K-split 4x for occupancy, LDS
// double-buffered with register prefetch (1 barrier / chunk), NT prefetch
// two chunks ahead, routing-weight end-scaling, global f32 atomic combine.
// grid = (ODIM/128, NEXP, KSPLIT), 256 threads = 8 waves.
// -------------------------------------------------------------------------
#define NBLK 128                 // output columns per block
#define KB   32                  // K chunk staged in LDS
#define SXP  (KB + 4)            // x-stage row pad: 36 floats, 16B aligned
#define KSPLIT 4
#define KSLICE (HDIM / KSPLIT)   // 1024 k-values per block
#define NCHUNK (KSLICE / KB)     // 32 chunks per block

__global__ __launch_bounds__(256) void moe_expert_kernel(
        const float* __restrict__ x,
        const float* __restrict__ expert_w,
        const float* __restrict__ w_gate,
        float* __restrict__ out) {
    const int e      = blockIdx.y;
    const int nbase  = blockIdx.x * NBLK;
    const int kbase0 = blockIdx.z * KSLICE;

    __shared__ float sW[2][KB][NBLK];   // 2 x 16 KB weight stage
    __shared__ float sX[2][NTOK][SXP];  // 2 x 4.5 KB activation stage
    __shared__ int   s_active;

    // Early exit if no token routed to this expert: skip all weight traffic.
    if (threadIdx.x == 0) s_active = 0;
    __syncthreads();
    if (threadIdx.x < NTOK) {
        if (w_gate[threadIdx.x * NEXP + e] != 0.0f) atomicOr(&s_active, 1);
    }
    __syncthreads();
    if (!s_active) return;

    const int t    = threadIdx.x;
    const int lane = t & 31;
    const int wave = t >> 5;              // 0..7 -> 16-col slice
    const int half = lane >> 4;           // 0 | 1
    const int lid  = lane & 15;
    const int ncol = nbase + wave * 16;

    // staging coordinates: each thread owns 4 W rows at a fixed 16B column,
    // and one 16B slice of the x chunk
    const int wrow = t >> 5;              // W stage row base (0..7)
    const int wcol = (t & 31) << 2;       // W stage column (float4 aligned)
    const int xrow = t >> 3;              // x stage row (0..31)
    const int xcol = (t & 7) << 2;        // x stage column

    const float* W = expert_w + (size_t)e * HDIM * ODIM;

    v8f acc0 = {};   // rows 0..15
    v8f acc1 = {};   // rows 16..31
    v4f rw0, rw1, rw2, rw3, rx;

    // ---- prefetch chunk 0 into registers, stage into buffer 0 ----
    {
        const int kb = kbase0;
        rw0 = *(const v4f*)(W + (size_t)(kb + wrow +  0) * ODIM + nbase + wcol);
        rw1 = *(const v4f*)(W + (size_t)(kb + wrow +  8) * ODIM + nbase + wcol);
        rw2 = *(const v4f*)(W + (size_t)(kb + wrow + 16) * ODIM + nbase + wcol);
        rw3 = *(const v4f*)(W + (size_t)(kb + wrow + 24) * ODIM + nbase + wcol);
        rx  = *(const v4f*)(x + (size_t)xrow * HDIM + kb + xcol);
        *(v4f*)(&sW[0][wrow +  0][wcol]) = rw0;
        *(v4f*)(&sW[0][wrow +  8][wcol]) = rw1;
        *(v4f*)(&sW[0][wrow + 16][wcol]) = rw2;
        *(v4f*)(&sW[0][wrow + 24][wcol]) = rw3;
        *(v4f*)(&sX[0][xrow][xcol])      = rx;
    }
    __syncthreads();

    for (int c = 0; c < NCHUNK; ++c) {
        const int cur = c & 1;
        const bool more = (c + 1) < NCHUNK;

        // ---- issue next chunk's global loads (latency hidden by WMMAs) ----
        if (more) {
            const int kb = kbase0 + (c + 1) * KB;
            rw0 = *(const v4f*)(W + (size_t)(kb + wrow +  0) * ODIM + nbase + wcol);
            rw1 = *(const v4f*)(W + (size_t)(kb + wrow +  8) * ODIM + nbase + wcol);
            rw2 = *(const v4f*)(W + (size_t)(kb + wrow + 16) * ODIM + nbase + wcol);
            rw3 = *(const v4f*)(W + (size_t)(kb + wrow + 24) * ODIM + nbase + wcol);
            rx  = *(const v4f*)(x + (size_t)xrow * HDIM + kb + xcol);
        }
        // ---- NT prefetch two chunks ahead (weights are streamed once) ----
        if (c + 2 < NCHUNK) {
            const int kb2 = kbase0 + (c + 2) * KB;
            __builtin_prefetch(W + (size_t)(kb2 + wrow +  0) * ODIM + nbase + wcol, 0, 0);
            __builtin_prefetch(W + (size_t)(kb2 + wrow +  8) * ODIM + nbase + wcol, 0, 0);
            __builtin_prefetch(W + (size_t)(kb2 + wrow + 16) * ODIM + nbase + wcol, 0, 0);
            __builtin_prefetch(W + (size_t)(kb2 + wrow + 24) * ODIM + nbase + wcol, 0, 0);
        }

        // ---- 8 WMMA steps over current chunk (K step 4) ----
#pragma unroll
        for (int kk = 0; kk < KB; kk += 4) {
            const int k0 = kk + half * 2;  // A: vgpr0 = K0|K2, vgpr1 = K1|K3
            v2f a0, a1, bf;
            a0.x = sX[cur][lid][k0];        a0.y = sX[cur][lid][k0 + 1];
            a1.x = sX[cur][lid + 16][k0];   a1.y = sX[cur][lid + 16][k0 + 1];
            bf.x = sW[cur][k0][wave * 16 + lid];
            bf.y = sW[cur][k0 + 1][wave * 16 + lid];
            acc0 = __builtin_amdgcn_wmma_f32_16x16x4_f32(
                       false, a0, false, bf, (short)0, acc0, false, false);
            acc1 = __builtin_amdgcn_wmma_f32_16x16x4_f32(
                       false, a1, false, bf, (short)0, acc1, false, false);
        }

        // ---- stage next chunk into the other buffer ----
        if (more) {
            const int nxt = cur ^ 1;
            *(v4f*)(&sW[nxt][wrow +  0][wcol]) = rw0;
            *(v4f*)(&sW[nxt][wrow +  8][wcol]) = rw1;
            *(v4f*)(&sW[nxt][wrow + 16][wcol]) = rw2;
            *(v4f*)(&sW[nxt][wrow + 24][wcol]) = rw3;
            *(v4f*)(&sX[nxt][xrow][xcol])      = rx;
        }
        __syncthreads();
    }

    // ---- apply routing weight per row and accumulate to output ----
    // C/D layout: VGPR r -> M=r (lanes 0-15) | M=r+8 (lanes 16-31), N = lid
#pragma unroll
    for (int r = 0; r < 8; ++r) {
        const int m0 = r + half * 8;        // row in tile 0
        const int m1 = m0 + 16;             // row in tile 1
        const float wg0 = w_gate[m0 * NEXP + e];
        const float wg1 = w_gate[m1 * NEXP + e];
        const int n = ncol + lid;
        if (wg0 != 0.0f) atomicAdd(&out[m0 * ODIM + n], acc0[r] * wg0);
        if (wg1 != 0.0f) atomicAdd(&out[m1 * ODIM + n], acc1[r] * wg1);
    }
}

// -------------------------------------------------------------------------
extern "C" void kernel_launch(void* const* d_in, const int* in_sizes, int n_in,
                              void* d_out, int out_size, void* d_ws, size_t ws_size,
                              hipStream_t stream) {
    const float* x        = (const float*)d_in[0];   // [32, 1, 4096]
    const float* gate_w   = (const float*)d_in[1];   // [4096, 8]
    const float* expert_w = (const float*)d_in[2];   // [8, 4096, 4096]
    float* out  = (float*)d_out;                     // [32, 1, 4096]
    float* w_ws = (float*)d_ws;                      // [32, 8] routing weights

    zero_out_kernel<<<dim3((out_size + 255) / 256), 256, 0, stream>>>(out, out_size);
    gate_topk_kernel<<<dim3(NTOK), 256, 0, stream>>>(x, gate_w, w_ws);
    moe_expert_kernel<<<dim3(ODIM / NBLK, NEXP, KSPLIT), 256, 0, stream>>>(
        x, expert_w, w_ws, out);
}